// CTCLayer_43654047596893
// MI455X (gfx1250) — compile-verified
//
#include <hip/hip_runtime.h>

#define B_    512
#define T_    512
#define C_    96
#define L_    64
#define S_    129          // 2*L+1
#define BLANK_ 95
#define EPS_  1e-7f
#define NEG_  -1.0e30f
#define NST   5            // states per lane: 32*5 = 160 >= 129

// ---- CDNA5 async global->LDS path (guarded; fallback = direct copy) ----
#if defined(__has_builtin)
# if __has_builtin(__builtin_amdgcn_global_load_async_to_lds_b128)
#  define HAVE_ASYNC_LDS 1
# endif
# if __has_builtin(__builtin_amdgcn_s_wait_asynccnt)
#  define HAVE_WAIT_ASYNC 1
# endif
#endif

#if defined(HAVE_WAIT_ASYNC)
# define WAIT_ASYNC(n) __builtin_amdgcn_s_wait_asynccnt(n)
#elif defined(HAVE_ASYNC_LDS)
# define WAIT_ASYNC(n) asm volatile("s_wait_asynccnt " #n ::: "memory")
#else
# define WAIT_ASYNC(n) ((void)0)
#endif

typedef int v4i_t __attribute__((vector_size(16)));
typedef __attribute__((address_space(1))) v4i_t* g_v4i_p;
typedef __attribute__((address_space(3))) v4i_t* l_v4i_p;

__device__ __forceinline__ void issue_row_fetch(const float* gsrc, float* ldst, int lane) {
#if defined(HAVE_ASYNC_LDS)
  // 96 floats = 384 B = 24 lanes x b128
  if (lane < 24) {
    __builtin_amdgcn_global_load_async_to_lds_b128(
        (g_v4i_p)(gsrc + lane * 4),
        (l_v4i_p)(ldst + lane * 4),
        0, 0);
  }
#else
  ldst[lane]      = gsrc[lane];
  ldst[lane + 32] = gsrc[lane + 32];
  ldst[lane + 64] = gsrc[lane + 64];
#endif
}

// One wave32 per batch element. 5 DP states per lane; cross-lane deps via shfl_up.
__global__ __launch_bounds__(32)
void ctc_forward_kernel(const int* __restrict__ y_true,
                        const float* __restrict__ y_pred,
                        float* __restrict__ loss_out)
{
  const int b    = blockIdx.x;
  const int lane = threadIdx.x;

  __shared__ __align__(16) float rowbuf[2][C_];
  __shared__ int   lab[L_];
  __shared__ float alpha_fin[NST * 32];

  // ---- labels: lab = where(y_true != -1, y_true, 0); lbl_len = count(!= -1) ----
  int yt0 = y_true[b * L_ + lane];
  int yt1 = y_true[b * L_ + 32 + lane];
  lab[lane]      = yt0 >= 0 ? yt0 : 0;
  lab[lane + 32] = yt1 >= 0 ? yt1 : 0;
  unsigned long long m0 = __ballot(yt0 != -1);
  unsigned long long m1 = __ballot(yt1 != -1);
  const int lbl_len = __popcll(m0) + __popcll(m1);
  __syncthreads();

  // ---- per-lane extended labels + can_skip (time-invariant) ----
  int  ec[NST];
  bool cs[NST];
  bool vld[NST];
  #pragma unroll
  for (int j = 0; j < NST; ++j) {
    const int s = lane * NST + j;
    vld[j] = (s < S_);
    if (s < S_ && (s & 1)) {
      const int li = (s - 1) >> 1;
      ec[j] = lab[li];                               // labels < BLANK always
      cs[j] = (s == 1) ? true : (ec[j] != lab[li - 1]);
    } else {
      ec[j] = BLANK_;
      cs[j] = false;
    }
  }

  const float* ybase = y_pred + (size_t)b * T_ * C_;
  float alpha[NST];

  issue_row_fetch(ybase, rowbuf[0], lane);           // prefetch row t=0

  for (int t = 0; t < T_; ++t) {
    const int  cur  = t & 1;
    const bool pref = (t + 1) < T_;
    if (pref) issue_row_fetch(ybase + (size_t)(t + 1) * C_, rowbuf[(t + 1) & 1], lane);
    if (pref) { WAIT_ASYNC(1); } else { WAIT_ASYNC(0); }   // row t is resident
    __syncthreads();

    float lpj[NST];
    #pragma unroll
    for (int j = 0; j < NST; ++j)
      lpj[j] = __logf(rowbuf[cur][ec[j]] + EPS_);

    if (t == 0) {
      #pragma unroll
      for (int j = 0; j < NST; ++j) {
        const int s = lane * NST + j;
        alpha[j] = (s <= 1) ? lpj[j] : NEG_;
      }
    } else {
      float up1 = __shfl_up(alpha[NST - 1], 1, 32);  // prev lane: s0-1
      float up2 = __shfl_up(alpha[NST - 2], 1, 32);  // prev lane: s0-2
      if (lane == 0) { up1 = NEG_; up2 = NEG_; }
      float na[NST];
      #pragma unroll
      for (int j = 0; j < NST; ++j) {
        const float a  = alpha[j];
        const float a1 = (j >= 1) ? alpha[j - 1] : up1;
        float       a2 = (j >= 2) ? alpha[j - 2] : ((j == 1) ? up1 : up2);
        a2 = cs[j] ? a2 : NEG_;
        const float m   = fmaxf(a, fmaxf(a1, a2));
        const float lse = m + __logf(__expf(a - m) + __expf(a1 - m) + __expf(a2 - m));
        na[j] = lse + lpj[j];
      }
      #pragma unroll
      for (int j = 0; j < NST; ++j) alpha[j] = vld[j] ? na[j] : NEG_;
    }
    __syncthreads();   // reads of rowbuf[cur] done before it is refilled at t+1
  }

  // ---- finish: -logaddexp(alpha[2*len], alpha[max(2*len-1,0)]) ----
  #pragma unroll
  for (int j = 0; j < NST; ++j) {
    const int s = lane * NST + j;
    if (s < S_) alpha_fin[s] = alpha[j];
  }
  __syncthreads();
  if (lane == 0) {
    const int ib = 2 * lbl_len;
    int il = ib - 1; if (il < 0) il = 0;
    const float ab = alpha_fin[ib];
    const float al = alpha_fin[il];
    const float m  = fmaxf(ab, al);
    loss_out[b] = -(m + __logf(__expf(ab - m) + __expf(al - m)));
  }
}

// ---- mean over 512 losses via accumulating V_WMMA_F32_16X16X4_F32 ----
// With B = ones, D[i][j] = rowsum(A[i]); summing every D entry over all 16
// columns counts each loss exactly 16 times -> mean = total / (16 * 512).
typedef float v2f __attribute__((ext_vector_type(2)));
typedef float v8f __attribute__((ext_vector_type(8)));

__global__ __launch_bounds__(32)
void ctc_mean_kernel(const float* __restrict__ losses, float* __restrict__ out)
{
  const int lane = threadIdx.x;
  v8f acc = {0.f, 0.f, 0.f, 0.f, 0.f, 0.f, 0.f, 0.f};
  v2f ones; ones.x = 1.0f; ones.y = 1.0f;
  #pragma unroll
  for (int ch = 0; ch < 8; ++ch) {          // 8 chunks x 64 losses (2/lane)
    v2f a;
    a.x = losses[ch * 64 + lane * 2 + 0];
    a.y = losses[ch * 64 + lane * 2 + 1];
    acc = __builtin_amdgcn_wmma_f32_16x16x4_f32(
        /*neg_a=*/false, a, /*neg_b=*/false, ones,
        /*c_mod=*/(short)0, acc, /*reuse_a=*/false, /*reuse_b=*/false);
  }
  float s = 0.f;
  #pragma unroll
  for (int k = 0; k < 8; ++k) s += acc[k];
  #pragma unroll
  for (int off = 16; off > 0; off >>= 1) s += __shfl_down(s, off, 32);
  if (lane == 0) out[0] = s * (1.0f / (16.0f * (float)B_));
}

extern "C" void kernel_launch(void* const* d_in, const int* in_sizes, int n_in,
                              void* d_out, int out_size, void* d_ws, size_t ws_size,
                              hipStream_t stream) {
  (void)in_sizes; (void)n_in; (void)out_size; (void)ws_size;
  const int*   y_true = (const int*)d_in[0];     // (B, L) int32
  const float* y_pred = (const float*)d_in[1];   // (B, T, C) float32
  float* losses = (float*)d_ws;                  // 512 floats scratch

  ctc_forward_kernel<<<B_, 32, 0, stream>>>(y_true, y_pred, losses);
  ctc_mean_kernel<<<1, 32, 0, stream>>>(losses, (float*)d_out);
}